// UnsupervisedFB_27659589386663
// MI455X (gfx1250) — compile-verified
//
#include <hip/hip_runtime.h>

#define BB    8
#define HH    512
#define WW    1024
#define HWP   (HH*WW)
#define TIL   16
#define HALO  2
#define REG   (TIL + 2*HALO)   // 20
#define MASK_COST 12.4f

#if defined(__has_builtin)
#  if __has_builtin(__builtin_amdgcn_global_load_async_to_lds_b32) && \
      __has_builtin(__builtin_amdgcn_s_wait_asynccnt)
#    define USE_ASYNC_LDS 1
#  endif
#endif

typedef float v2f __attribute__((ext_vector_type(2)));
typedef float v8f __attribute__((ext_vector_type(8)));

#ifdef USE_ASYNC_LDS
typedef __attribute__((address_space(1))) int* gas_int_t;   // global (AS1)
typedef __attribute__((address_space(3))) int* las_int_t;   // LDS    (AS3)
#endif

__device__ __forceinline__ float sigm(float x){ return 1.0f / (1.0f + __expf(-x)); }

// taps = [-1/12, 2/3, 0, -2/3, 1/12] at indices 0..4, zero outside (branchless)
__device__ __forceinline__ float tapw(int d){
  return (d == 0) ? -1.0f/12.0f :
         (d == 1) ?  2.0f/3.0f  :
         (d == 3) ? -2.0f/3.0f  :
         (d == 4) ?  1.0f/12.0f : 0.0f;
}

// bilinear sample with border-clamped indices, weights from unclamped coords
__device__ __forceinline__ float bilin_plane(const float* __restrict__ plane, float gx, float gy){
  float x0f = floorf(gx), y0f = floorf(gy);
  float ax = gx - x0f,  ay = gy - y0f;
  int x0 = (int)x0f, y0 = (int)y0f;
  int x0i = min(max(x0,     0), WW-1);
  int x1i = min(max(x0 + 1, 0), WW-1);
  int y0i = min(max(y0,     0), HH-1);
  int y1i = min(max(y0 + 1, 0), HH-1);
  float v00 = plane[y0i*WW + x0i];
  float v01 = plane[y0i*WW + x1i];
  float v10 = plane[y1i*WW + x0i];
  float v11 = plane[y1i*WW + x1i];
  return (1.0f-ay)*((1.0f-ax)*v00 + ax*v01) + ay*((1.0f-ax)*v10 + ax*v11);
}

__global__ __launch_bounds__(256) void fb_energy_kernel(
    const float* __restrict__ flowf, const float* __restrict__ flowb,
    const float* __restrict__ img1,  const float* __restrict__ img2,
    const float* __restrict__ target, float* __restrict__ partial)
{
  __shared__ float sFlow[2*REG*REG];   // flowA, tile+halo
  __shared__ float sDiff[3*REG*REG];   // imgA - warp(imgB, flowA), tile+halo (zero pad OOB)
  __shared__ float sC[TIL*TIL];        // sum of squared gradients per interior pixel
  __shared__ float sRedE[256];
  __shared__ float sRedP[256];

  const int tid   = threadIdx.x;
  const int b     = blockIdx.z;
  const int tileX = blockIdx.x * TIL;
  const int tileY = blockIdx.y * TIL;

  // WMMA role constants (wave-uniform where needed)
  const int wv     = tid >> 5;          // wave id 0..7
  const int lane   = tid & 31;
  const int hl     = lane >> 4;         // lane half
  const int ln     = lane & 15;
  const int ch     = wv >> 1;           // channel 0..2 (waves 0..5)
  const int orient = wv & 1;            // 0 = horizontal, 1 = vertical

  // Precompute band-matrix (tap) operand pairs: same for both orientations,
  // both directions, all tiles.  A/B layout: lane-half holds K = {2h, 2h+1}.
  float tw0[5], tw1[5];
  #pragma unroll
  for (int t = 0; t < 5; ++t){
    int k0 = 4*t + 2*hl;
    tw0[t] = tapw(k0     - ln);
    tw1[t] = tapw(k0 + 1 - ln);
  }

  float accE = 0.0f, accP = 0.0f;

  for (int dir = 0; dir < 2; ++dir){
    const float* fa = dir ? flowb : flowf;   // flow being evaluated
    const float* fo = dir ? flowf : flowb;   // other flow (warped)
    const float* ia = dir ? img2  : img1;
    const float* ib = dir ? img1  : img2;

    __syncthreads();  // protect LDS reuse across directions

    // ---- Phase 1: stage flowA (tile + halo) into LDS; prefetch gather sources ----
    for (int i = tid; i < REG*REG; i += 256){
      int ry = i / REG, rx = i - ry*REG;
      int gyy = tileY - HALO + ry;
      int gxx = tileX - HALO + rx;
      int cy = min(max(gyy, 0), HH-1);
      int cx = min(max(gxx, 0), WW-1);
      __builtin_prefetch(fo + (size_t)b*2*HWP + cy*WW + cx, 0, 1);   // global_prefetch_b8
      __builtin_prefetch(ib + (size_t)b*3*HWP + cy*WW + cx, 0, 1);
      bool inb = (gyy >= 0) && (gyy < HH) && (gxx >= 0) && (gxx < WW);
#ifdef USE_ASYNC_LDS
      // CDNA5 async LDS DMA, tracked by ASYNCcnt (OOB halo cells are never read)
      if (inb){
        const float* g0 = fa + (size_t)b*2*HWP + gyy*WW + gxx;
        __builtin_amdgcn_global_load_async_to_lds_b32(
            (gas_int_t)g0,         (las_int_t)&sFlow[i],           0, 0);
        __builtin_amdgcn_global_load_async_to_lds_b32(
            (gas_int_t)(g0 + HWP), (las_int_t)&sFlow[REG*REG + i], 0, 0);
      }
#else
      float v0 = 0.0f, v1 = 0.0f;
      if (inb){
        const float* g0 = fa + (size_t)b*2*HWP + gyy*WW + gxx;
        v0 = g0[0];
        v1 = g0[HWP];
      }
      sFlow[i]           = v0;
      sFlow[REG*REG + i] = v1;
#endif
    }
    sC[tid] = 0.0f;
#ifdef USE_ASYNC_LDS
    __builtin_amdgcn_s_wait_asynccnt(0);
#endif
    __syncthreads();

    // ---- Phase 2: warped diff region E = imgA - warp(imgB, flowA) (zero outside image) ----
    for (int i = tid; i < REG*REG; i += 256){
      int ry = i / REG, rx = i - ry*REG;
      int gyy = tileY - HALO + ry;
      int gxx = tileX - HALO + rx;
      float D0 = 0.0f, D1 = 0.0f, D2 = 0.0f;
      if (gyy >= 0 && gyy < HH && gxx >= 0 && gxx < WW){
        float fx = sFlow[i], fy = sFlow[REG*REG + i];
        float gxw = (float)gxx + fx, gyw = (float)gyy + fy;
        const float* ib0 = ib + (size_t)b*3*HWP;
        float w0 = bilin_plane(ib0,         gxw, gyw);
        float w1 = bilin_plane(ib0 +   HWP, gxw, gyw);
        float w2 = bilin_plane(ib0 + 2*HWP, gxw, gyw);
        const float* ia0 = ia + (size_t)b*3*HWP + gyy*WW + gxx;
        D0 = ia0[0]     - w0;
        D1 = ia0[HWP]   - w1;
        D2 = ia0[2*HWP] - w2;
      }
      sDiff[i]             = D0;
      sDiff[REG*REG + i]   = D1;
      sDiff[2*REG*REG + i] = D2;
    }
    __syncthreads();

    // ---- Phase 3: 5-tap gradients as chained f32 WMMA (exact banded matmul) ----
    // GH(16x16) = E(16x20) x Wh(20x16), Wh[K][n] = tap(K-n)
    // GV(16x16) = Wv(16x20) x E(20x16), Wv[m][K] = tap(K-m)
    if (wv < 6){                        // wave-uniform branch; EXEC all-1s inside
      const float* E = &sDiff[ch*REG*REG];
      v8f acc = {0,0,0,0,0,0,0,0};
      #pragma unroll
      for (int t = 0; t < 5; ++t){
        int k0     = 4*t + 2*hl;
        int r0     = orient ? k0          : (ln + HALO);
        int c0     = orient ? (ln + HALO) : k0;
        int stride = orient ? REG         : 1;
        int a0     = r0*REG + c0;
        v2f Dv; Dv.x = E[a0]; Dv.y = E[a0 + stride];
        v2f Tv; Tv.x = tw0[t]; Tv.y = tw1[t];
        v2f Av = orient ? Tv : Dv;      // vertical:   taps x data
        v2f Bv = orient ? Dv : Tv;      // horizontal: data x taps
        acc = __builtin_amdgcn_wmma_f32_16x16x4_f32(
            false, Av, false, Bv, (short)0, acc, false, false);
      }
      #pragma unroll
      for (int r = 0; r < 8; ++r){
        int m = r + 8*hl;               // D layout: VGPR r -> M=r (lanes 0-15) / M=r+8 (16-31)
        atomicAdd(&sC[m*TIL + ln], acc[r]*acc[r]);   // ds_add_f32
      }
    }
    __syncthreads();

    // ---- Phase 4: per-pixel energy terms ----
    {
      int tx = tid & 15, ty = tid >> 4;
      int gx = tileX + tx, gy = tileY + ty;
      int rC = (ty + HALO)*REG + (tx + HALO);
      float fx = sFlow[rC], fy = sFlow[REG*REG + rC];
      float Xc = (float)gx + fx, Yc = (float)gy + fy;
      float bmask = sigm(Xc + 0.5f) * sigm(((float)WW - 0.5f) - Xc)
                  * sigm(Yc + 0.5f) * sigm(((float)HH - 0.5f) - Yc);
      const float* fo0 = fo + (size_t)b*2*HWP;
      float wbx = bilin_plane(fo0,       Xc, Yc);
      float wby = bilin_plane(fo0 + HWP, Xc, Yc);
      float mag = fx*fx + fy*fy + wbx*wbx + wby*wby;
      float sx = fx + wbx, sy = fy + wby;
      float d2 = sx*sx + sy*sy;
      float occ = sigm((0.01f*mag + 0.5f) - d2);
      float mask = bmask * occ;
      float d0 = sDiff[rC], d1 = sDiff[REG*REG + rC], dq = sDiff[2*REG*REG + rC];
      float Aterm = d0*d0 + d1*d1 + dq*dq;
      float Bm = 0.0f;
      if (gx < WW-1){ float ex = sFlow[rC+1]   - fx, ey = sFlow[REG*REG + rC+1]   - fy; Bm += ex*ex + ey*ey; }
      if (gy < HH-1){ float ex = sFlow[rC+REG] - fx, ey = sFlow[REG*REG + rC+REG] - fy; Bm += ex*ex + ey*ey; }
      float Cterm = sC[ty*TIL + tx];
      accE += sqrtf(Aterm + 1e-5f)*mask        // data
            + sqrtf(Cterm + 1e-5f)*mask        // gradient
            + MASK_COST*(1.0f - mask)          // mask cost
            + sqrtf(d2 + 1e-5f)*mask           // fwd-bwd consistency
            + sqrtf(Bm + 1e-5f);               // smoothness
      if (dir == 0){
        const float* tg = target + (size_t)b*2*HWP + gy*WW + gx;
        float t0 = fx - tg[0], t1 = fy - tg[HWP];
        accP += sqrtf(t0*t0 + t1*t1);
      }
    }
  } // dir

  // ---- Block reduction (deterministic) ----
  __syncthreads();
  sRedE[tid] = accE; sRedP[tid] = accP;
  __syncthreads();
  for (int s = 128; s > 0; s >>= 1){
    if (tid < s){ sRedE[tid] += sRedE[tid + s]; sRedP[tid] += sRedP[tid + s]; }
    __syncthreads();
  }
  if (tid == 0){
    int bl = (b * gridDim.y + blockIdx.y) * gridDim.x + blockIdx.x;
    partial[2*bl]     = sRedE[0];
    partial[2*bl + 1] = sRedP[0];
  }
}

__global__ __launch_bounds__(256) void fb_reduce_kernel(
    const float* __restrict__ partial, int n, float* __restrict__ out)
{
  __shared__ float sE[256], sP[256];
  float e = 0.0f, p = 0.0f;
  for (int i = threadIdx.x; i < n; i += 256){ e += partial[2*i]; p += partial[2*i + 1]; }
  sE[threadIdx.x] = e; sP[threadIdx.x] = p;
  __syncthreads();
  for (int s = 128; s > 0; s >>= 1){
    if (threadIdx.x < s){ sE[threadIdx.x] += sE[threadIdx.x + s]; sP[threadIdx.x] += sP[threadIdx.x + s]; }
    __syncthreads();
  }
  if (threadIdx.x == 0){
    out[0] = sE[0] / (float)BB;                    // mean(energy) over batch
    out[1] = sP[0] / (float)((size_t)BB*HH*WW);    // mean EPE over B,H,W
  }
}

extern "C" void kernel_launch(void* const* d_in, const int* in_sizes, int n_in,
                              void* d_out, int out_size, void* d_ws, size_t ws_size,
                              hipStream_t stream)
{
  (void)in_sizes; (void)n_in; (void)out_size; (void)ws_size;
  const float* flowf  = (const float*)d_in[0];
  const float* flowb  = (const float*)d_in[1];
  const float* img1   = (const float*)d_in[2];
  const float* img2   = (const float*)d_in[3];
  const float* target = (const float*)d_in[4];
  float* out     = (float*)d_out;
  float* partial = (float*)d_ws;   // 2 floats per block = 128 KB

  dim3 grid(WW/TIL, HH/TIL, BB);   // 64 x 32 x 8 = 16384 tiles
  fb_energy_kernel<<<grid, 256, 0, stream>>>(flowf, flowb, img1, img2, target, partial);
  int nblk = (WW/TIL)*(HH/TIL)*BB;
  fb_reduce_kernel<<<1, 256, 0, stream>>>(partial, nblk, out);
}